// BSplineWarp_11433202942708
// MI455X (gfx1250) — compile-verified
//
#include <hip/hip_runtime.h>
#include <hip/hip_bf16.h>

typedef __attribute__((ext_vector_type(2))) float v2f;
typedef __attribute__((ext_vector_type(8))) float v8f;

#define IMG 1024
#define CTRL 32
#define NBATCH 16

__device__ __forceinline__ int clampi(int v, int lo, int hi) {
    return v < lo ? lo : (v > hi ? hi : v);
}

// Keys cubic-convolution weight, a = -0.75, matching the reference's where(d<=1, near, where(d<2, far, 0))
__device__ __forceinline__ float keys_w(float d) {
    const float A_ = -0.75f;
    float wn = ((A_ + 2.0f) * d - (A_ + 3.0f)) * d * d + 1.0f;
    float wf = A_ * (((d - 5.0f) * d + 8.0f) * d - 4.0f);
    return (d <= 1.0f) ? wn : ((d < 2.0f) ? wf : 0.0f);
}

// Build composite weight matrix W[1024][32]:
// W[y][a] = sum_{dy in {-1,0,1}} s(dy) * KeysCubicWeight(clamp(y+dy), a)  (with control-index clamping)
// s = (1/6, 4/6, 1/6) is the B-spline smoothing at integer positions (frac == 0).
__global__ void bsw_build_w(float* __restrict__ W) {
    int tid = blockIdx.x * blockDim.x + threadIdx.x;
    if (tid >= IMG * CTRL) return;
    int a = tid & (CTRL - 1);
    int y = tid >> 5;
    float acc = 0.0f;
#pragma unroll
    for (int dy = -1; dy <= 1; ++dy) {
        int p = clampi(y + dy, 0, IMG - 1);
        float posf = ((float)p + 0.5f) * (1.0f / 32.0f) - 0.5f;
        float fi = floorf(posf);
        float t = posf - fi;
        int i0 = (int)fi;
        float s = (dy == 0) ? (4.0f / 6.0f) : (1.0f / 6.0f);
#pragma unroll
        for (int j = 0; j < 4; ++j) {   // taps at i0-1 .. i0+2, offsets -1..2
            int idx = clampi(i0 + j - 1, 0, CTRL - 1);
            float d = fabsf(t - (float)(j - 1));
            acc += (idx == a) ? s * keys_w(d) : 0.0f;
        }
    }
    W[tid] = acc;
}

// Stage 1: tmp[n,c,y,b] = sum_a W[y][a] * disp[n,c,a,b].  1M outputs, 32 MACs each. L2-resident.
__global__ void bsw_stage1(const float* __restrict__ disp, const float* __restrict__ W,
                           float* __restrict__ tmp) {
    int tid = blockIdx.x * blockDim.x + threadIdx.x;  // < 16*2*1024*32
    int b  = tid & (CTRL - 1);
    int y  = (tid >> 5) & (IMG - 1);
    int nc = tid >> 15;
    const float* Wrow = W + y * CTRL;
    const float* drow = disp + nc * (CTRL * CTRL);
    float acc = 0.0f;
#pragma unroll
    for (int a = 0; a < CTRL; ++a)
        acc = fmaf(Wrow[a], drow[a * CTRL + b], acc);
    tmp[tid] = acc;
}

// Stage 2: per-wave 16x16 pixel tile, both channels, via V_WMMA_F32_16X16X4_F32 (K=32 in 8 steps).
// D[m][n'] = sum_b tmp[nc][y0+m][b] * W[x0+n'][b]; then out = img + {c0,c1}.
__global__ void bsw_stage2(const float* __restrict__ img, const float* __restrict__ tmp,
                           const float* __restrict__ W, float* __restrict__ out) {
    int wave = threadIdx.x >> 5;
    int lane = threadIdx.x & 31;
    int tile = blockIdx.x * (blockDim.x >> 5) + wave;   // 16 * 64 * 64 = 65536 tiles
    int n  = tile >> 12;
    int ty = (tile >> 6) & 63;
    int tx = tile & 63;
    int y0 = ty << 4, x0 = tx << 4;
    int m    = lane & 15;   // A: row M; B: column N; C: column N
    int half = lane >> 4;   // selects K pair within a 4-wide K step

    const float* tA0 = tmp + ((size_t)((n * 2 + 0) * IMG) + y0 + m) * CTRL;
    const float* tA1 = tmp + ((size_t)((n * 2 + 1) * IMG) + y0 + m) * CTRL;
    const float* wB  = W + (size_t)(x0 + m) * CTRL;

    v8f acc0 = {};
    v8f acc1 = {};
#pragma unroll
    for (int k = 0; k < 8; ++k) {
        int kb = k * 4 + half * 2;
        v2f a0 = *(const v2f*)(tA0 + kb);   // A frag: A[m][kb], A[m][kb+1]
        v2f a1 = *(const v2f*)(tA1 + kb);
        v2f b  = *(const v2f*)(wB + kb);    // B frag: B[kb][m], B[kb+1][m] == W[x0+m][kb..kb+1]
        acc0 = __builtin_amdgcn_wmma_f32_16x16x4_f32(false, a0, false, b, (short)0, acc0, false, false);
        acc1 = __builtin_amdgcn_wmma_f32_16x16x4_f32(false, a1, false, b, (short)0, acc1, false, false);
    }

    // C/D layout: VGPR r -> row (r + 8*half), column m. Store interleaved {c0,c1} float2 (coalesced).
#pragma unroll
    for (int r = 0; r < 8; ++r) {
        int y = y0 + r + half * 8;
        int x = x0 + m;
        size_t pix = (((size_t)n * IMG + y) * IMG + x);
        v2f p = *(const v2f*)(img + pix * 2);
        v2f o;
        o.x = p.x + acc0[r];
        o.y = p.y + acc1[r];
        *(v2f*)(out + pix * 2) = o;
    }
}

extern "C" void kernel_launch(void* const* d_in, const int* in_sizes, int n_in,
                              void* d_out, int out_size, void* d_ws, size_t ws_size,
                              hipStream_t stream) {
    const float* img  = (const float*)d_in[0];   // [16,1024,1024,2] f32
    const float* disp = (const float*)d_in[1];   // [16,2,32,32] f32
    float* out = (float*)d_out;                  // [16,1024,1024,2] f32

    float* W   = (float*)d_ws;                   // 1024*32 floats   = 128 KB
    float* tmp = W + IMG * CTRL;                 // 16*2*1024*32 f32 = 4 MiB

    // Stage 0: composite (bicubic-upsample ∘ integer-grid cubic B-spline) weights.
    bsw_build_w<<<(IMG * CTRL + 255) / 256, 256, 0, stream>>>(W);

    // Stage 1: contract control-grid y dimension. tmp stays hot in L2.
    bsw_stage1<<<(NBATCH * 2 * IMG * CTRL) / 256, 256, 0, stream>>>(disp, W, tmp);

    // Stage 2: contract x dimension with f32 WMMA, fuse add of image_coordinates.
    // 65536 tiles, 4 waves (one tile each) per 128-thread block.
    bsw_stage2<<<65536 / 4, 128, 0, stream>>>(img, tmp, W, out);
}